// Atoms_12266426598008
// MI455X (gfx1250) — compile-verified
//
#include <hip/hip_runtime.h>
#include <hip/hip_bf16.h>
#include <math.h>

// ---------------- problem constants ----------------
#define BATCH   64
#define CH      128
#define NFRAMES 128
#define NSAMP   32768
#define NHARM   16
#define NTONE   17          // 1 fundamental + 16 harmonics
#define NNOTES  64
#define NFILT   256
#define CHUNK   256         // samples per frame period (NSAMP / NFRAMES)
#define NCHUNK  128

#define PI_F 3.14159265358979323846f
#define INV_SQRT_N 0.005524271728019903f  // 1/sqrt(32768)

typedef __attribute__((ext_vector_type(2))) float v2f;
typedef __attribute__((ext_vector_type(8))) float v8f;

__device__ __forceinline__ float leakyf(float v)   { return v >= 0.f ? v : 0.2f * v; }
__device__ __forceinline__ float sigmoidf(float v) { return 1.f / (1.f + expf(-v)); }

// F.interpolate(linear, align_corners=False) coefficients, in_size=128, out=32768
__device__ __forceinline__ void interp_coeff(int n, int& i0, int& i1, float& w) {
    float pos = ((float)n + 0.5f) * (1.0f / 256.0f) - 0.5f;
    pos = fminf(fmaxf(pos, 0.f), 127.f);
    float fp = floorf(pos);
    i0 = (int)fp;
    i1 = min(i0 + 1, 127);
    w = pos - fp;
}

// ---------------- WMMA f32 GEMM: C[M,N] = act(A[M,K] @ W[K,N] + bias) ----------------
// One wave per 16x16 tile. grid = (M/16, N/16), block = 32 (full wave, EXEC all-ones).
// act: 0=none, 1=leaky, 2=sigmoid
__global__ __launch_bounds__(32) void gemm_wmma(const float* __restrict__ A,
                                                const float* __restrict__ W,
                                                const float* __restrict__ bias,
                                                float* __restrict__ C,
                                                int K, int N, int act) {
    const int mt   = blockIdx.x;
    const int nt   = blockIdx.y;
    const int lane = threadIdx.x;
    const int r    = lane & 15;   // M row (A) / N col (B,C)
    const int half = lane >> 4;   // K split for A/B frags, M split for C/D

    v8f acc = {0.f, 0.f, 0.f, 0.f, 0.f, 0.f, 0.f, 0.f};
    const float* Arow = A + (size_t)(mt * 16 + r) * K;
    const int col = nt * 16 + r;

    for (int k0 = 0; k0 < K; k0 += 4) {
        const int ka = k0 + half * 2;
        v2f a, b;
        a.x = Arow[ka];
        a.y = Arow[ka + 1];
        b.x = W[(size_t)ka * N + col];
        b.y = W[(size_t)(ka + 1) * N + col];
        acc = __builtin_amdgcn_wmma_f32_16x16x4_f32(false, a, false, b,
                                                    (short)0, acc, false, false);
    }
    const float bv = bias[col];
#pragma unroll
    for (int i = 0; i < 8; ++i) {
        float v = acc[i] + bv;
        if (act == 1)      v = leakyf(v);
        else if (act == 2) v = sigmoidf(v);
        C[(size_t)(mt * 16 + half * 8 + i) * N + col] = v;
    }
}

// ---------------- ConvTranspose1d k=4 s=2 p=1 via WMMA ----------------
// For a fixed output position t, out[:, :, t] is a sum over (at most) two taps of
// X[:, :, j] @ W[:, :, k]  (shapes [B,CH] @ [CH,CH]) -> tiled 16x16 WMMA GEMM.
// Tap validity depends only on blockIdx.z => wave-uniform branches, EXEC stays all-ones.
// X:[B][CH][Lin]  Wt:[in=CH][out=CH][4]  Y:[B][CH][2*Lin]   grid=(B/16, CH/16, Lout)
__global__ __launch_bounds__(32) void convt_wmma(const float* __restrict__ X,
                                                 const float* __restrict__ Wt,
                                                 const float* __restrict__ bias,
                                                 float* __restrict__ Y, int Lin) {
    const int mt   = blockIdx.x;       // batch tile
    const int nt   = blockIdx.y;       // out-channel tile
    const int t    = blockIdx.z;       // output position
    const int Lout = Lin * 2;
    const int lane = threadIdx.x;
    const int r    = lane & 15;
    const int half = lane >> 4;
    const int col  = nt * 16 + r;

    v8f acc = {0.f, 0.f, 0.f, 0.f, 0.f, 0.f, 0.f, 0.f};
#pragma unroll
    for (int k = 0; k < 4; ++k) {
        const int m = t + k - 2;              // dilated-input index
        if (m < 0 || (m & 1)) continue;
        const int j = m >> 1;
        if (j >= Lin) continue;
        const float* Xb = X + (size_t)(mt * 16 + r) * CH * Lin + j;   // row b, tap j
        for (int k0 = 0; k0 < CH; k0 += 4) {
            const int ka = k0 + half * 2;
            v2f a, b;
            a.x = Xb[(size_t)ka * Lin];
            a.y = Xb[(size_t)(ka + 1) * Lin];
            b.x = Wt[((size_t)ka * CH + col) * 4 + k];
            b.y = Wt[((size_t)(ka + 1) * CH + col) * 4 + k];
            acc = __builtin_amdgcn_wmma_f32_16x16x4_f32(false, a, false, b,
                                                        (short)0, acc, false, false);
        }
    }
    const float bv = bias[col];
#pragma unroll
    for (int i = 0; i < 8; ++i) {
        const int row = mt * 16 + half * 8 + i;    // batch index
        Y[((size_t)row * CH + col) * Lout + t] = leakyf(acc[i] + bv);
    }
}

// ---------------- gumbel-hard f0 selection ----------------
__global__ void f0_select(const float* __restrict__ logits, const float* __restrict__ gum,
                          const float* __restrict__ cf, float* __restrict__ f0) {
    __shared__ float sv[NNOTES];
    __shared__ int   si[NNOTES];
    const int b = blockIdx.x, n = threadIdx.x;
    float u = gum[b * NNOTES + n];
    sv[n] = logits[b * NNOTES + n] + (-logf(-logf(u)));
    si[n] = n;
    __syncthreads();
    for (int s = NNOTES / 2; s > 0; s >>= 1) {
        if (n < s && sv[n + s] > sv[n]) { sv[n] = sv[n + s]; si[n] = si[n + s]; }
        __syncthreads();
    }
    if (n == 0) f0[b] = cf[si[0]];
}

// ---------------- out conv k=3 pad=1 (CH -> 1) + bias + head activation ----------------
// act: 3=tanh, 2=sigmoid.  X:[B][CH][128]  Wo:[CH][3]  D:[B][128]
__global__ void out_conv(const float* __restrict__ X, const float* __restrict__ Wo,
                         const float* __restrict__ ob, float* __restrict__ D, int act) {
    int idx = blockIdx.x * blockDim.x + threadIdx.x;
    if (idx >= BATCH * NFRAMES) return;
    const int t = idx % NFRAMES;
    const int b = idx / NFRAMES;
    float acc = ob[0];
    for (int i = 0; i < CH; ++i) {
        const float* xr = X + (size_t)(b * CH + i) * NFRAMES;
        float s = Wo[i * 3 + 1] * xr[t];
        if (t > 0)           s = fmaf(Wo[i * 3 + 0], xr[t - 1], s);
        if (t < NFRAMES - 1) s = fmaf(Wo[i * 3 + 2], xr[t + 1], s);
        acc += s;
    }
    D[idx] = (act == 3) ? tanhf(acc) : sigmoidf(acc);
}

// ---------------- frame-rate tones + amp-with-decay ----------------
__global__ void build_frames(const float* __restrict__ f0sel, const float* __restrict__ f0ch,
                             const float* __restrict__ ampF, const float* __restrict__ decayF,
                             const float* __restrict__ factors,
                             float* __restrict__ tonesF, float* __restrict__ awdF) {
    int idx = blockIdx.x * blockDim.x + threadIdx.x;
    if (idx >= BATCH * NTONE * NFRAMES) return;
    const int f = idx % NFRAMES;
    const int c = (idx / NFRAMES) % NTONE;
    const int b = idx / (NFRAMES * NTONE);

    const float f0   = f0sel[b];
    const float freq = f0 * (1.f + 0.1f * f0ch[b * NFRAMES + f]);   // f0 + f0*FREQ_CHANGE*tanh
    const float mult = (c == 0) ? 1.f : (1.f + (float)(c - 1) * (16.f / 15.f)); // linspace(1,17,16)
    float tone = freq * mult;
    if (tone >= 1.f) tone = 0.f;                                    // zero super-nyquist
    tonesF[idx] = tone;

    const float fac = (c == 0) ? 1.f : factors[b * NHARM + (c - 1)];
    const float a   = fac * ampF[b * NFRAMES + f];
    float awd = a;
    if (f > 0) awd = a + fac * ampF[b * NFRAMES + f - 1] * decayF[b * NFRAMES + f];
    awdF[idx] = awd;
}

// ---------------- chunked phase scan: per-(b,c,chunk) sums of interp(tone)*pi ----------
__global__ void chunk_sums(const float* __restrict__ tonesF, float* __restrict__ pfx) {
    int idx = blockIdx.x * blockDim.x + threadIdx.x;
    if (idx >= BATCH * NTONE * NCHUNK) return;
    const int j  = idx % NCHUNK;
    const int bc = idx / NCHUNK;
    const float* tf = tonesF + (size_t)bc * NFRAMES;
    float s = 0.f;
    const int n0 = j * CHUNK;
    for (int n = 0; n < CHUNK; ++n) {
        int i0, i1; float w;
        interp_coeff(n0 + n, i0, i1, w);
        s += (tf[i0] * (1.f - w) + tf[i1] * w) * PI_F;
    }
    pfx[idx] = s;
}

__global__ void prefix_scan(float* __restrict__ pfx) {
    int bc = blockIdx.x * blockDim.x + threadIdx.x;
    if (bc >= BATCH * NTONE) return;
    float* p = pfx + (size_t)bc * NCHUNK;
    float run = 0.f;
    for (int j = 0; j < NCHUNK; ++j) { float t = p[j]; p[j] = run; run += t; }
}

// ---------------- noise filter: phases of first 256 samples, then fixed-order channel sum
__global__ void filt_phase(const float* __restrict__ tonesF, float* __restrict__ ph0) {
    int bc = blockIdx.x * blockDim.x + threadIdx.x;
    if (bc >= BATCH * NTONE) return;
    const float* tf = tonesF + (size_t)bc * NFRAMES;
    float* out = ph0 + (size_t)bc * NFILT;
    float phase = 0.f;
    for (int n = 0; n < NFILT; ++n) {
        int i0, i1; float w;
        interp_coeff(n, i0, i1, w);
        phase += (tf[i0] * (1.f - w) + tf[i1] * w) * PI_F;
        out[n] = phase;
    }
}

__global__ void filt_sum(const float* __restrict__ ph0, float* __restrict__ filtS) {
    int idx = blockIdx.x * blockDim.x + threadIdx.x;
    if (idx >= BATCH * NFILT) return;
    const int n = idx % NFILT;
    const int b = idx / NFILT;
    float s = 0.f;
    for (int c = 0; c < NTONE; ++c)
        s += sinf(ph0[((size_t)b * NTONE + c) * NFILT + n]);
    filtS[idx] = s;
}

// ---------------- circular conv of noise with summed 256-tap filter -> d_out (scratch) --
__global__ void noise_conv(const float* __restrict__ noise, const float* __restrict__ filtS,
                           float* __restrict__ out) {
    int idx = blockIdx.x * blockDim.x + threadIdx.x;
    if (idx >= BATCH * NSAMP) return;
    const int n = idx & (NSAMP - 1);
    const int b = idx >> 15;
    const float* nb = noise + (size_t)b * NSAMP;
    const float* fb = filtS + (size_t)b * NFILT;
    float s = 0.f;
    for (int k = 0; k < NFILT; ++k)
        s = fmaf(fb[k], nb[(n - k) & (NSAMP - 1)], s);
    out[idx] = s * INV_SQRT_N;   // ortho-norm rfft/irfft scaling
}

// ---------------- final combine: sum_c sin(phase_c)*awd_c * mix + bl*(1-mix) -----------
__global__ void combine(const float* __restrict__ tonesF, const float* __restrict__ awdF,
                        const float* __restrict__ mixF, const float* __restrict__ pfx,
                        float* __restrict__ out) {
    int idx = blockIdx.x * blockDim.x + threadIdx.x;
    if (idx >= BATCH * NCHUNK) return;
    const int j = idx % NCHUNK;
    const int b = idx / NCHUNK;

    float phase[NTONE];
    const float* tB = tonesF + (size_t)b * NTONE * NFRAMES;
    const float* aB = awdF   + (size_t)b * NTONE * NFRAMES;
    const float* pB = pfx    + (size_t)b * NTONE * NCHUNK;
    const float* mB = mixF   + (size_t)b * NFRAMES;
#pragma unroll
    for (int c = 0; c < NTONE; ++c) phase[c] = pB[c * NCHUNK + j];

    const int n0 = j * CHUNK;
    for (int nn = 0; nn < CHUNK; ++nn) {
        const int n = n0 + nn;
        int i0, i1; float w;
        interp_coeff(n, i0, i1, w);
        const float w0 = 1.f - w;
        const float mixv = mB[i0] * w0 + mB[i1] * w;
        float acc = 0.f;
#pragma unroll
        for (int c = 0; c < NTONE; ++c) {
            const float tone = tB[c * NFRAMES + i0] * w0 + tB[c * NFRAMES + i1] * w;
            phase[c] += tone * PI_F;
            const float awdv = aB[c * NFRAMES + i0] * w0 + aB[c * NFRAMES + i1] * w;
            acc = fmaf(sinf(phase[c]), awdv, acc);
        }
        const float bl = out[(size_t)b * NSAMP + n];
        out[(size_t)b * NSAMP + n] = acc * mixv + bl * (1.f - mixv);
    }
}

// ======================================================================================
extern "C" void kernel_launch(void* const* d_in, const int* in_sizes, int n_in,
                              void* d_out, int out_size, void* d_ws, size_t ws_size,
                              hipStream_t stream) {
    (void)in_sizes; (void)n_in; (void)out_size; (void)ws_size;
    auto F = [&](int i) { return (const float*)d_in[i]; };

    // Input map (recursive insertion-order flatten of setup_inputs dict):
    // 0:x  1-3:f0.ws  4-6:f0.bs  7:f0.wf  8:f0.bf
    // 9-11:spec.ws 12-14:spec.bs 15:spec.wf 16:spec.bf
    // heads (init_w,init_b, w0,b0,w1,b1,w2,b2,w3,b3, out_w,out_b):
    //   f0_change@17  amp@29  decay@41  mix@53
    // 65:center_freqs  66:gumbel  67:noise
    const float* x     = F(0);
    const float* cf    = F(65);
    const float* gum   = F(66);
    const float* noise = F(67);

    float* ws = (float*)d_ws;
    float* hA      = ws;                       // 8192
    float* hB      = hA + 8192;                // 8192
    float* logits  = hB + 8192;                // 4096
    float* factors = logits + 4096;            // 1024
    float* f0sel   = factors + 1024;           // 64
    float* convA   = f0sel + 64;               // 1048576
    float* convB   = convA + (size_t)BATCH * CH * NFRAMES;  // 1048576
    float* f0chF   = convB + (size_t)BATCH * CH * NFRAMES;  // 8192
    float* ampF    = f0chF + BATCH * NFRAMES;  // 8192
    float* decayF  = ampF + BATCH * NFRAMES;   // 8192
    float* mixF    = decayF + BATCH * NFRAMES; // 8192
    float* tonesF  = mixF + BATCH * NFRAMES;   // 139264
    float* awdF    = tonesF + (size_t)BATCH * NTONE * NFRAMES;  // 139264
    float* pfx     = awdF + (size_t)BATCH * NTONE * NFRAMES;    // 139264
    float* ph0     = pfx + (size_t)BATCH * NTONE * NCHUNK;      // 278528
    float* filtS   = ph0 + (size_t)BATCH * NTONE * NFILT;       // 16384
    float* out     = (float*)d_out;

    const dim3 g128(4, 8), g64(4, 4), g16(4, 1), g1024(4, 64);

    // f0 lin stack -> logits
    gemm_wmma<<<g128, 32, 0, stream>>>(x,  F(1), F(4), hA, CH, CH, 1);
    gemm_wmma<<<g128, 32, 0, stream>>>(hA, F(2), F(5), hB, CH, CH, 1);
    gemm_wmma<<<g128, 32, 0, stream>>>(hB, F(3), F(6), hA, CH, CH, 1);
    gemm_wmma<<<g64,  32, 0, stream>>>(hA, F(7), F(8), logits, CH, NNOTES, 0);
    // spec lin stack -> sigmoid factors
    gemm_wmma<<<g128, 32, 0, stream>>>(x,  F(9),  F(12), hA, CH, CH, 1);
    gemm_wmma<<<g128, 32, 0, stream>>>(hA, F(10), F(13), hB, CH, CH, 1);
    gemm_wmma<<<g128, 32, 0, stream>>>(hB, F(11), F(14), hA, CH, CH, 1);
    gemm_wmma<<<g16,  32, 0, stream>>>(hA, F(15), F(16), factors, CH, NHARM, 2);

    f0_select<<<BATCH, NNOTES, 0, stream>>>(logits, gum, cf, f0sel);

    // four upsampler heads (init GEMM + 4x WMMA conv-transpose + out conv)
    struct Head { int base; float* dst; int act; };
    const Head heads[4] = { {17, f0chF, 3}, {29, ampF, 2}, {41, decayF, 2}, {53, mixF, 2} };
    for (int h = 0; h < 4; ++h) {
        const int base = heads[h].base;
        gemm_wmma<<<g1024, 32, 0, stream>>>(x, F(base), F(base + 1), convA, CH, CH * 8, 0);
        int Lin = 8;
        float* src = convA; float* dst = convB;
        for (int l = 0; l < 4; ++l) {
            convt_wmma<<<dim3(4, 8, Lin * 2), 32, 0, stream>>>(
                src, F(base + 2 + 2 * l), F(base + 3 + 2 * l), dst, Lin);
            Lin *= 2;
            float* tmp = src; src = dst; dst = tmp;
        }
        out_conv<<<(BATCH * NFRAMES + 255) / 256, 256, 0, stream>>>(
            src, F(base + 10), F(base + 11), heads[h].dst, heads[h].act);
    }

    build_frames<<<(BATCH * NTONE * NFRAMES + 255) / 256, 256, 0, stream>>>(
        f0sel, f0chF, ampF, decayF, factors, tonesF, awdF);

    chunk_sums<<<(BATCH * NTONE * NCHUNK + 255) / 256, 256, 0, stream>>>(tonesF, pfx);
    prefix_scan<<<(BATCH * NTONE + 63) / 64, 64, 0, stream>>>(pfx);

    filt_phase<<<(BATCH * NTONE + 63) / 64, 64, 0, stream>>>(tonesF, ph0);
    filt_sum<<<(BATCH * NFILT + 255) / 256, 256, 0, stream>>>(ph0, filtS);

    noise_conv<<<(BATCH * NSAMP + 255) / 256, 256, 0, stream>>>(noise, filtS, out);
    combine<<<(BATCH * NCHUNK + 63) / 64, 64, 0, stream>>>(tonesF, awdF, mixF, pfx, out);
}